// ChebshevGCNN_61452392071699
// MI455X (gfx1250) — compile-verified
//
#include <hip/hip_runtime.h>

// Problem constants (fixed by the reference)
#define B_   32
#define N_   8192
#define F_   16
#define E_   65536
#define O_   32
#define NF_  (N_ * F_)                    // 131072 floats per (batch) term
#define ONF_ ((size_t)O_ * NF_)           // per-batch output stride
#define BNF_ ((size_t)B_ * NF_)           // 4,194,304 floats per Chebyshev term

typedef __attribute__((ext_vector_type(2))) float v2f;
typedef __attribute__((ext_vector_type(4))) float f4;
typedef __attribute__((ext_vector_type(8))) float v8f;

// ---------------------------------------------------------------------------
// init: dst = -src (fold the "- T_{k-2}" of the recurrence into the
// accumulator) or dst = 0 for the first SpMM. One element per thread.
// ---------------------------------------------------------------------------
__global__ void cheb_init(float* __restrict__ dst, const float* __restrict__ src) {
    size_t i = (size_t)blockIdx.x * blockDim.x + threadIdx.x;
    dst[i] = src ? -src[i] : 0.0f;
}

// ---------------------------------------------------------------------------
// SpMM scatter: out[b, rows[e], f] += scale * vals[e] * in[b, cols[e], f]
// One thread per (edge, feature). 16 consecutive lanes share one edge, so
// row/col/val loads are uniform per half-wave and the x-gather is coalesced.
// Terms live in L2 (80 MB << 192 MB), so atomics land in L2, not HBM.
// ---------------------------------------------------------------------------
__global__ void cheb_spmm(const int* __restrict__ rows, const int* __restrict__ cols,
                          const float* __restrict__ vals, const float* __restrict__ xin,
                          float* __restrict__ xout, float scale) {
    size_t idx = (size_t)blockIdx.x * blockDim.x + threadIdx.x;
    int    f   = (int)(idx & 15);
    size_t eg  = idx >> 4;                 // global edge id in [0, B*E)
    int    b   = (int)(eg >> 16);          // E = 65536 = 2^16
    size_t eb  = (size_t)b * E_ + (eg & 65535);

    int   r = rows[eb];
    int   c = cols[eb];
    float v = vals[eb] * scale;

    size_t base = (size_t)b * NF_;
    float contrib = v * xin[base + (size_t)c * F_ + f];
    unsafeAtomicAdd(&xout[base + (size_t)r * F_ + f], contrib);  // global_atomic_add_f32
}

// ---------------------------------------------------------------------------
// Dense combine with V_WMMA_F32_16X16X4_F32:
//   y[b,o,n,f] = relu( sum_k T_k[b,n,f] * w[k,o] + bias[o] )
// One wave handles one (b, n): A tile = 16 f-rows x 5 k (padded to 8, two
// WMMAs), two o-tiles of 16 reusing A. D lanes map to 8 contiguous f-values
// -> two nontemporal float4 stores per lane per o-tile (512 MB write-once).
// ---------------------------------------------------------------------------
__global__ void __launch_bounds__(256) cheb_combine(
        const float* __restrict__ x,  const float* __restrict__ t1,
        const float* __restrict__ t2, const float* __restrict__ t3,
        const float* __restrict__ t4, const float* __restrict__ w,
        const float* __restrict__ bias, float* __restrict__ out) {
    int wave = blockIdx.x * 8 + (threadIdx.x >> 5);   // 262144 waves total
    int lane = threadIdx.x & 31;
    int b    = wave >> 13;                            // / 8192
    int n    = wave & 8191;
    int m    = lane & 15;                             // A: f-row   B/D: o-col
    int hi   = lane >> 4;                             // lane-half selects K pair

    // ---- A matrix (16x4 f32): lanes 0-15 hold K={0,1}, lanes 16-31 K={2,3}
    size_t tidx = (size_t)b * NF_ + (size_t)n * F_ + m;   // T_k[b, n, f=m]
    float v0 = x[tidx],  v1 = t1[tidx], v2 = t2[tidx];
    float v3 = t3[tidx], v4 = t4[tidx];
    v2f a0, a1;
    a0.x = hi ? v2 : v0;          // K = 0 / 2
    a0.y = hi ? v3 : v1;          // K = 1 / 3
    a1.x = hi ? 0.0f : v4;        // K = 4 (k=5..7 are zero padding)
    a1.y = 0.0f;

    size_t obase0 = (size_t)b * ONF_ + (size_t)n * F_ + (size_t)(hi * 8);

#pragma unroll
    for (int ot = 0; ot < 2; ++ot) {
        int o = ot * 16 + m;                           // B/D column = output chan
        float w0 = w[0 * O_ + o], w1 = w[1 * O_ + o], w2 = w[2 * O_ + o];
        float w3 = w[3 * O_ + o], w4 = w[4 * O_ + o];
        v2f b0, b1;
        b0.x = hi ? w2 : w0;
        b0.y = hi ? w3 : w1;
        b1.x = hi ? 0.0f : w4;
        b1.y = 0.0f;

        v8f c = {};
        c = __builtin_amdgcn_wmma_f32_16x16x4_f32(false, a0, false, b0,
                                                  (short)0, c, false, false);
        c = __builtin_amdgcn_wmma_f32_16x16x4_f32(false, a1, false, b1,
                                                  (short)0, c, false, false);

        float bv = bias[o];
        f4 lo, hv;
        lo.x = fmaxf(c[0] + bv, 0.0f);
        lo.y = fmaxf(c[1] + bv, 0.0f);
        lo.z = fmaxf(c[2] + bv, 0.0f);
        lo.w = fmaxf(c[3] + bv, 0.0f);
        hv.x = fmaxf(c[4] + bv, 0.0f);
        hv.y = fmaxf(c[5] + bv, 0.0f);
        hv.z = fmaxf(c[6] + bv, 0.0f);
        hv.w = fmaxf(c[7] + bv, 0.0f);

        // lane's 8 results are f = hi*8 + {0..7}: contiguous, 32B aligned
        float* p = out + obase0 + (size_t)o * NF_;
        __builtin_nontemporal_store(lo, (f4*)p);
        __builtin_nontemporal_store(hv, (f4*)(p + 4));
    }
}

// ---------------------------------------------------------------------------
extern "C" void kernel_launch(void* const* d_in, const int* in_sizes, int n_in,
                              void* d_out, int out_size, void* d_ws, size_t ws_size,
                              hipStream_t stream) {
    const float* x    = (const float*)d_in[0];
    const int*   rows = (const int*)  d_in[1];
    const int*   cols = (const int*)  d_in[2];
    const float* vals = (const float*)d_in[3];
    const float* w    = (const float*)d_in[4];
    const float* bias = (const float*)d_in[5];
    float*       out  = (float*)d_out;

    // Workspace: T1..T4, 16 MB each (T0 == x)
    float* t1 = (float*)d_ws;
    float* t2 = t1 + BNF_;
    float* t3 = t2 + BNF_;
    float* t4 = t3 + BNF_;

    dim3 blk(256);
    const int initBlocks = (int)(BNF_ / 256);                  // 16384
    const int spmmBlocks = (int)(((size_t)B_ * E_ * F_) / 256); // 131072
    const int combBlocks = (B_ * N_) / 8;                      // 32768 (8 waves/blk)

    // T1 = L x
    cheb_init<<<initBlocks, blk, 0, stream>>>(t1, nullptr);
    cheb_spmm<<<spmmBlocks, blk, 0, stream>>>(rows, cols, vals, x,  t1, 1.0f);
    // T2 = 2 L T1 - T0
    cheb_init<<<initBlocks, blk, 0, stream>>>(t2, x);
    cheb_spmm<<<spmmBlocks, blk, 0, stream>>>(rows, cols, vals, t1, t2, 2.0f);
    // T3 = 2 L T2 - T1
    cheb_init<<<initBlocks, blk, 0, stream>>>(t3, t1);
    cheb_spmm<<<spmmBlocks, blk, 0, stream>>>(rows, cols, vals, t2, t3, 2.0f);
    // T4 = 2 L T3 - T2
    cheb_init<<<initBlocks, blk, 0, stream>>>(t4, t2);
    cheb_spmm<<<spmmBlocks, blk, 0, stream>>>(rows, cols, vals, t3, t4, 2.0f);

    // y = relu(einsum + bias) via f32 WMMA
    cheb_combine<<<combBlocks, blk, 0, stream>>>(x, t1, t2, t3, t4, w, bias, out);
}